// dgl_HNN_43379169689779
// MI455X (gfx1250) — compile-verified
//
#include <hip/hip_runtime.h>

typedef __attribute__((ext_vector_type(2))) float v2f;
typedef __attribute__((ext_vector_type(8))) float v8f;

#define IN_F  16
#define HID_F 64

// ---------------------------------------------------------------- utilities
__global__ void zero_kernel(float* __restrict__ p, long n) {
  long i = (long)blockIdx.x * blockDim.x + threadIdx.x;
  long stride = (long)gridDim.x * blockDim.x;
  for (; i < n; i += stride) p[i] = 0.0f;
}

__global__ void degree_kernel(const int* __restrict__ src, const int* __restrict__ dst,
                              float* __restrict__ deg_out, float* __restrict__ deg_in,
                              int n_edges) {
  int e = blockIdx.x * blockDim.x + threadIdx.x;
  if (e < n_edges) {
    atomicAdd(&deg_out[src[e]], 1.0f);
    atomicAdd(&deg_in[dst[e]], 1.0f);
  }
}

__global__ void rsqrt_kernel(float* __restrict__ r_out, float* __restrict__ r_in, int n) {
  int i = blockIdx.x * blockDim.x + threadIdx.x;
  if (i < n) {
    r_out[i] = rsqrtf(fmaxf(r_out[i], 1.0f));
    r_in[i]  = rsqrtf(fmaxf(r_in[i], 1.0f));
  }
}

// ------------------------------------------------- edge gather + scatter-add
// agg[dst] += h[src] * r_out[src]; F features, one thread per (edge, 4-feature group)
template <int F>
__global__ void scatter_kernel(const float* __restrict__ h, const int* __restrict__ src,
                               const int* __restrict__ dst, const float* __restrict__ r_out,
                               float* __restrict__ agg, int n_edges) {
  constexpr int GROUPS = F / 4;
  long tid = (long)blockIdx.x * blockDim.x + threadIdx.x;
  long e = tid / GROUPS;
  int  g = (int)(tid % GROUPS) * 4;
  if (e < n_edges) {
    int s = src[e], d = dst[e];
    float r = r_out[s];
    const float4 v = *(const float4*)(h + (long)s * F + g);   // global_load_b128
    float* o = agg + (long)d * F + g;
    atomicAdd(o + 0, v.x * r);
    atomicAdd(o + 1, v.y * r);
    atomicAdd(o + 2, v.z * r);
    atomicAdd(o + 3, v.w * r);
  }
}

// ---------------------------------------------------------- WMMA node stages
// y1 = tanh( (agg1 * r_in) @ W1 + b1 )   : [N,16] @ [16,64]
// one wave per 16-node tile; 4 N-tiles, K=16 in 4 steps of V_WMMA_F32_16X16X4_F32
__global__ void node1_wmma(const float* __restrict__ agg1, const float* __restrict__ r_in,
                           const float* __restrict__ w1, const float* __restrict__ b1,
                           float* __restrict__ y1, int n_tiles, int n_nodes) {
  int wave = (int)(((long)blockIdx.x * blockDim.x + threadIdx.x) >> 5);
  int lane = threadIdx.x & 31;
  if (wave >= n_tiles) return;

  int row = wave * 16 + (lane & 15);
  int rowc = row < n_nodes ? row : n_nodes - 1;           // clamp (keeps EXEC all-1s)
  int khalf = (lane >> 4) * 2;                            // lanes 16-31 hold K+2,K+3
  float rin = r_in[rowc];
  const float* Arow = agg1 + (long)rowc * IN_F;

  v2f afrag[4];
#pragma unroll
  for (int ks = 0; ks < 4; ++ks) {
    int k = ks * 4 + khalf;
    afrag[ks].x = Arow[k] * rin;
    afrag[ks].y = Arow[k + 1] * rin;
  }

  int nloc = lane & 15;
  int mbase = wave * 16 + ((lane >> 4) ? 8 : 0);          // C/D: vgpr r -> M = r (+8 hi half)
#pragma unroll
  for (int nt = 0; nt < 4; ++nt) {
    int n = nt * 16 + nloc;
    v8f c = {};
#pragma unroll
    for (int ks = 0; ks < 4; ++ks) {
      int k = ks * 4 + khalf;
      v2f b;
      b.x = w1[k * HID_F + n];
      b.y = w1[(k + 1) * HID_F + n];
      c = __builtin_amdgcn_wmma_f32_16x16x4_f32(false, afrag[ks], false, b,
                                                (short)0, c, false, false);
    }
    float bias = b1[n];
#pragma unroll
    for (int r = 0; r < 8; ++r) {
      int m = mbase + r;
      if (m < n_nodes) y1[(long)m * HID_F + n] = tanhf(c[r] + bias);
    }
  }
}

// out = ((agg2 * r_in) @ W2 + b2) @ J^T  : [N,64] @ [64,16], fused J permutation
__global__ void node2_wmma(const float* __restrict__ agg2, const float* __restrict__ r_in,
                           const float* __restrict__ w2, const float* __restrict__ b2,
                           float* __restrict__ out, int n_tiles, int n_nodes) {
  int wave = (int)(((long)blockIdx.x * blockDim.x + threadIdx.x) >> 5);
  int lane = threadIdx.x & 31;
  if (wave >= n_tiles) return;

  int row = wave * 16 + (lane & 15);
  int rowc = row < n_nodes ? row : n_nodes - 1;
  int khalf = (lane >> 4) * 2;
  float rin = r_in[rowc];
  const float* Arow = agg2 + (long)rowc * HID_F;
  int n = lane & 15;

  v8f c = {};
#pragma unroll
  for (int ks = 0; ks < 16; ++ks) {                       // K = 64 in 16 steps
    int k = ks * 4 + khalf;
    v2f a, b;
    a.x = Arow[k] * rin;
    a.y = Arow[k + 1] * rin;
    b.x = w2[k * IN_F + n];
    b.y = w2[(k + 1) * IN_F + n];
    c = __builtin_amdgcn_wmma_f32_16x16x4_f32(false, a, false, b,
                                              (short)0, c, false, false);
  }

  float bias = b2[n];
  // J^T fusion: y2 column j -> out column (j+8)&15, sign + for j>=8, - for j<8
  int ocol = (n + 8) & 15;
  float sign = (n >= 8) ? 1.0f : -1.0f;
  int mbase = wave * 16 + ((lane >> 4) ? 8 : 0);
#pragma unroll
  for (int r = 0; r < 8; ++r) {
    int m = mbase + r;
    if (m < n_nodes) out[(long)m * IN_F + ocol] = sign * (c[r] + bias);
  }
}

// ------------------------------------------------------------------ launcher
extern "C" void kernel_launch(void* const* d_in, const int* in_sizes, int n_in,
                              void* d_out, int out_size, void* d_ws, size_t ws_size,
                              hipStream_t stream) {
  const float* x  = (const float*)d_in[0];
  const int*   src = (const int*)d_in[1];
  const int*   dst = (const int*)d_in[2];
  const float* w1 = (const float*)d_in[3];
  const float* b1 = (const float*)d_in[4];
  const float* w2 = (const float*)d_in[5];
  const float* b2 = (const float*)d_in[6];
  float* out = (float*)d_out;

  int n_nodes = in_sizes[0] / IN_F;
  int n_edges = in_sizes[1];
  int n_tiles = (n_nodes + 15) / 16;

  float* ws   = (float*)d_ws;
  float* rOut = ws;                       // [N]
  float* rIn  = ws + (long)n_nodes;       // [N]
  float* agg1 = ws + 2L  * n_nodes;       // [N,16]
  float* y1   = ws + 18L * n_nodes;       // [N,64]
  float* agg2 = ws + 82L * n_nodes;       // [N,64]
  long total_ws = 146L * n_nodes;

  // 1. zero degree accumulators + aggregation buffers (ws is poisoned)
  zero_kernel<<<2048, 256, 0, stream>>>(ws, total_ws);

  // 2. degrees
  degree_kernel<<<(n_edges + 255) / 256, 256, 0, stream>>>(src, dst, rOut, rIn, n_edges);

  // 3. deg^{-1/2} with clip(.,1)
  rsqrt_kernel<<<(n_nodes + 255) / 256, 256, 0, stream>>>(rOut, rIn, n_nodes);

  // 4. conv1 edge scatter: agg1[dst] += x[src]*rOut[src]
  {
    long t = (long)n_edges * (IN_F / 4);
    scatter_kernel<IN_F><<<(unsigned)((t + 255) / 256), 256, 0, stream>>>(
        x, src, dst, rOut, agg1, n_edges);
  }

  // 5. conv1 dense stage (WMMA) + tanh
  node1_wmma<<<(n_tiles * 32 + 255) / 256, 256, 0, stream>>>(
      agg1, rIn, w1, b1, y1, n_tiles, n_nodes);

  // 6. conv2 edge scatter: agg2[dst] += y1[src]*rOut[src]
  {
    long t = (long)n_edges * (HID_F / 4);
    scatter_kernel<HID_F><<<(unsigned)((t + 255) / 256), 256, 0, stream>>>(
        y1, src, dst, rOut, agg2, n_edges);
  }

  // 7. conv2 dense stage (WMMA) + bias + J^T permutation
  node2_wmma<<<(n_tiles * 32 + 255) / 256, 256, 0, stream>>>(
      agg2, rIn, w2, b2, out, n_tiles, n_nodes);
}